// CrfDecoder_16252156248443
// MI455X (gfx1250) — compile-verified
//
#include <hip/hip_runtime.h>
#include <hip/hip_bf16.h>
#include <math.h>

// CRF decoder: out[b,c] = numerator(b,c) - denominator(b,c)
// T=512, B=256, C=4, K=64
constexpr int Tn = 512;
constexpr int Bn = 256;
constexpr int Cn = 4;
constexpr int Kn = 64;
constexpr int PITCH = 68;   // LDS row pitch in floats (bank-skew: +4)

typedef float v2f __attribute__((ext_vector_type(2)));
typedef float v8f __attribute__((ext_vector_type(8)));

#define LN2F 0.6931471805599453f

// ---------------------------------------------------------------------------
// Kernel 1: numerator (path score). One wave per (b,c) pair; lanes split T.
// Writes out[b*C+c] = numerator.
// ---------------------------------------------------------------------------
__global__ __launch_bounds__(256) void crf_numerator_kernel(
    const float* __restrict__ emis,      // [T,B,C,K]
    const int*   __restrict__ tags,      // [T,B,C]
    const int*   __restrict__ tok,       // [B]
    const float* __restrict__ trans,     // [C,K,K]
    const float* __restrict__ startT,    // [C,K]
    const float* __restrict__ endT,      // [C,K]
    float*       __restrict__ out)       // [B,C]
{
    const int lane = threadIdx.x & 31;
    const int wid  = (blockIdx.x * (blockDim.x >> 5)) + (threadIdx.x >> 5);
    if (wid >= Bn * Cn) return;
    const int b = wid >> 2;
    const int c = wid & 3;

    int len = tok[b];
    len = len < 1 ? 1 : (len > Tn ? Tn : len);

    float acc = 0.0f;
    for (int t = 1 + lane; t < Tn; t += 32) {
        if (t < len) {
            const int tp = tags[((t - 1) * Bn + b) * Cn + c];
            const int tc = tags[(t * Bn + b) * Cn + c];
            acc += trans[(c * Kn + tp) * Kn + tc]
                 + emis[((t * Bn + b) * Cn + c) * Kn + tc];
        }
    }
    #pragma unroll
    for (int off = 16; off > 0; off >>= 1)
        acc += __shfl_xor(acc, off, 32);

    if (lane == 0) {
        const int t0   = tags[(0 * Bn + b) * Cn + c];
        const int tlst = tags[((len - 1) * Bn + b) * Cn + c];
        acc += startT[c * Kn + t0]
             + emis[((0 * Bn + b) * Cn + c) * Kn + t0]
             + endT[c * Kn + tlst];
        out[b * Cn + c] = acc;
    }
}

// ---------------------------------------------------------------------------
// Kernel 2: forward algorithm (denominator) via WMMA f32 16x16x4.
// Grid: 64 blocks = 16 batch-tiles x 4 components. Block: 128 threads (4 waves).
// Alpha state lives in registers in the WMMA D-tile layout (lane holds 8
// (row,col) elements). Row-max: shfl_xor over 16-lane col groups + 16x4 LDS
// cross-wave combine. exp(alpha-m) is written directly to the skewed sExp
// tile consumed as WMMA A fragments. Length masking is a branchless cndmask.
// Subtracts denominator from out[b,c] (numerator written by kernel 1).
// ---------------------------------------------------------------------------
__global__ __launch_bounds__(128) void crf_forward_kernel(
    const float* __restrict__ emis,      // [T,B,C,K]
    const int*   __restrict__ tok,       // [B]
    const float* __restrict__ trans,     // [C,K,K]
    const float* __restrict__ startT,    // [C,K]
    const float* __restrict__ endT,      // [C,K]
    float*       __restrict__ out)       // [B,C]
{
    __shared__ float sExp[16][PITCH];                 // exp(alpha - rowmax)
    __shared__ alignas(16) float sPmax[16][4];        // per-wave row-max partials
    __shared__ alignas(16) float sPsum[16][4];        // per-wave row-sum partials

    const int tid  = threadIdx.x;
    const int lane = tid & 31;
    const int wave = tid >> 5;

    const int c  = blockIdx.x & 3;
    const int b0 = (blockIdx.x >> 2) * 16;
    const int n0 = wave * 16;                 // this wave's N-tile base

    // --- constant B fragments: exp(transitions[c, k, n]) in WMMA B layout ---
    // B is KxN (4x16): lanes 0-15 -> K offsets {0,1}, lanes 16-31 -> {2,3};
    // lane%16 selects the N column.
    const int halfsel = (lane < 16) ? 0 : 2;
    const int nn = n0 + (lane & 15);
    v2f bfrag[16];
    #pragma unroll
    for (int kb = 0; kb < 16; ++kb) {
        const int k = kb * 4 + halfsel;
        bfrag[kb].x = __expf(trans[(c * Kn + k)     * Kn + nn]);
        bfrag[kb].y = __expf(trans[(c * Kn + k + 1) * Kn + nn]);
    }

    // D-tile coordinates: VGPR j -> row M = j + rowLo, col = colD.
    const int rowLo = (lane < 16) ? 0 : 8;
    const int colD  = n0 + (lane & 15);
    const int ar    = lane & 15;              // A-fragment row (M) for this lane

    // Sequence lengths for this lane's 8 rows (registers).
    int lenR[8];
    #pragma unroll
    for (int j = 0; j < 8; ++j) {
        int L = tok[b0 + rowLo + j];
        lenR[j] = L < 1 ? 1 : (L > Tn ? Tn : L);
    }

    // alpha0 = start_transitions + emissions[0], register-resident.
    float alphaReg[8];
    {
        const float sT = startT[c * Kn + colD];
        #pragma unroll
        for (int j = 0; j < 8; ++j)
            alphaReg[j] = sT
                + emis[((0 * Bn + (b0 + rowLo + j)) * Cn + c) * Kn + colD];
    }

    // Prefetch emissions for t=1 (register double buffer).
    float eC[8];
    #pragma unroll
    for (int j = 0; j < 8; ++j)
        eC[j] = emis[((1 * Bn + (b0 + rowLo + j)) * Cn + c) * Kn + colD];

    for (int t = 1; t < Tn; ++t) {
        // Issue next-step emission loads early (latency hidden behind phases).
        float eN[8];
        if (t + 1 < Tn) {
            #pragma unroll
            for (int j = 0; j < 8; ++j)
                eN[j] = emis[(((t + 1) * Bn + (b0 + rowLo + j)) * Cn + c) * Kn + colD];
        }

        // (a) per-row max: shfl reduce over this wave's 16 columns.
        float m[8];
        #pragma unroll
        for (int j = 0; j < 8; ++j) {
            float p = alphaReg[j];
            p = fmaxf(p, __shfl_xor(p, 1, 16));
            p = fmaxf(p, __shfl_xor(p, 2, 16));
            p = fmaxf(p, __shfl_xor(p, 4, 16));
            p = fmaxf(p, __shfl_xor(p, 8, 16));
            m[j] = p;
        }
        if ((lane & 15) == 0) {
            #pragma unroll
            for (int j = 0; j < 8; ++j) sPmax[rowLo + j][wave] = m[j];
        }
        __syncthreads();
        #pragma unroll
        for (int j = 0; j < 8; ++j) {
            const float4 p4 = *(const float4*)sPmax[rowLo + j];  // ds_load_b128
            m[j] = fmaxf(fmaxf(p4.x, p4.y), fmaxf(p4.z, p4.w));
        }

        // (b) sExp = exp(alpha - rowmax), written straight from registers.
        #pragma unroll
        for (int j = 0; j < 8; ++j)
            sExp[rowLo + j][colD] = __expf(alphaReg[j] - m[j]);
        __syncthreads();

        // (c) beta = sExp(16x64) @ expT(64x16) via 16 WMMAs (K step 4).
        v8f acc = {};
        #pragma unroll
        for (int kb = 0; kb < 16; ++kb) {
            v2f a;
            a.x = sExp[ar][kb * 4 + halfsel];
            a.y = sExp[ar][kb * 4 + halfsel + 1];
            acc = __builtin_amdgcn_wmma_f32_16x16x4_f32(
                false, a, false, bfrag[kb], (short)0, acc, false, false);
        }

        // (d) alpha' = log(beta) + rowmax + emis[t], branchless length mask.
        #pragma unroll
        for (int j = 0; j < 8; ++j) {
            const float nv = __builtin_amdgcn_logf(acc[j]) * LN2F + m[j] + eC[j];
            alphaReg[j] = (t < lenR[j]) ? nv : alphaReg[j];
        }
        #pragma unroll
        for (int j = 0; j < 8; ++j) eC[j] = eN[j];
    }

    // Final: denominator = logsumexp(alpha + end_transitions); subtract.
    {
        const float ev = endT[c * Kn + colD];
        float m[8], s[8];
        #pragma unroll
        for (int j = 0; j < 8; ++j) {
            const float v = alphaReg[j] + ev;
            float p = v;
            p = fmaxf(p, __shfl_xor(p, 1, 16));
            p = fmaxf(p, __shfl_xor(p, 2, 16));
            p = fmaxf(p, __shfl_xor(p, 4, 16));
            p = fmaxf(p, __shfl_xor(p, 8, 16));
            m[j] = p;
            float e = __expf(v - p);
            e += __shfl_xor(e, 1, 16);
            e += __shfl_xor(e, 2, 16);
            e += __shfl_xor(e, 4, 16);
            e += __shfl_xor(e, 8, 16);
            s[j] = e;
        }
        if ((lane & 15) == 0) {
            #pragma unroll
            for (int j = 0; j < 8; ++j) {
                sPmax[rowLo + j][wave] = m[j];
                sPsum[rowLo + j][wave] = s[j];
            }
        }
        __syncthreads();
        if (tid < 16) {
            const float4 pm = *(const float4*)sPmax[tid];
            const float4 ps = *(const float4*)sPsum[tid];
            const float M = fmaxf(fmaxf(pm.x, pm.y), fmaxf(pm.z, pm.w));
            const float S = ps.x * __expf(pm.x - M) + ps.y * __expf(pm.y - M)
                          + ps.z * __expf(pm.z - M) + ps.w * __expf(pm.w - M);
            const float den = __builtin_amdgcn_logf(S) * LN2F + M;
            out[(b0 + tid) * Cn + c] -= den;
        }
    }
}

// ---------------------------------------------------------------------------
extern "C" void kernel_launch(void* const* d_in, const int* in_sizes, int n_in,
                              void* d_out, int out_size, void* d_ws, size_t ws_size,
                              hipStream_t stream) {
    const float* emis   = (const float*)d_in[0];  // emissions [T,B,C,K]
    const int*   tags   = (const int*)  d_in[1];  // tags      [T,B,C]
    const int*   tok    = (const int*)  d_in[2];  // token_sizes [B]
    const float* trans  = (const float*)d_in[3];  // transitions [C,K,K]
    const float* startT = (const float*)d_in[4];  // start_transitions [C,K]
    const float* endT   = (const float*)d_in[5];  // end_transitions   [C,K]
    float* out = (float*)d_out;                   // [B,C]

    // Kernel 1: numerator -> out (pure write). 1024 waves, 8 waves/block.
    crf_numerator_kernel<<<(Bn * Cn) / 8, 256, 0, stream>>>(
        emis, tags, tok, trans, startT, endT, out);

    // Kernel 2: forward algorithm, subtracts denominator from out.
    crf_forward_kernel<<<16 * Cn, 128, 0, stream>>>(
        emis, tok, trans, startT, endT, out);

    (void)d_ws; (void)ws_size; (void)in_sizes; (void)n_in; (void)out_size;
}